// Net_43267500540651
// MI455X (gfx1250) — compile-verified
//
#include <hip/hip_runtime.h>

// Problem constants (from reference)
#define BATCH 16
#define SEQ   2048
#define HID   1024
#define CH    15          // real channels
#define CHP   16          // padded channels for 16x16 WMMA tile
#define NW    800         // NUM_WORDS
#define NTOK  (BATCH * SEQ)   // 32768 tokens
#define NSEG  (BATCH * NW)    // 12800 segments

typedef __attribute__((ext_vector_type(2))) float v2f;
typedef __attribute__((ext_vector_type(8))) float v8f;

// ---------------------------------------------------------------------------
// Kernel 0: zero segment sums + counts in workspace (d_ws is poisoned).
// ---------------------------------------------------------------------------
__global__ __launch_bounds__(256) void zero_ws_kernel(float* __restrict__ sums,
                                                      float* __restrict__ counts) {
  int i = blockIdx.x * 256 + threadIdx.x;
  if (i < NSEG * CHP) sums[i] = 0.0f;
  if (i < NSEG)       counts[i] = 0.0f;
}

// ---------------------------------------------------------------------------
// Kernel 1: y = x @ W^T via V_WMMA_F32_16X16X4_F32, fused with segment
// scatter-add (atomics) of the 16-channel logits + segment counts.
// One wave -> one 16-token x 16-channel tile, K-loop over H=1024 in steps of 4.
// Block = 256 threads = 8 waves = 8 consecutive tiles.
// ---------------------------------------------------------------------------
__global__ __launch_bounds__(256) void gemm_scatter_kernel(
    const float* __restrict__ x,      // [NTOK, HID]
    const int*   __restrict__ wid,    // [NTOK]
    const float* __restrict__ W,      // [CH, HID]
    float* __restrict__ sums,         // [NSEG, CHP]
    float* __restrict__ counts) {     // [NSEG]
  // Stage W into LDS, transposed to [k][c] (exactly 64KB, conflict-free reads).
  __shared__ float ldsW[HID * CHP];
  for (int idx = threadIdx.x; idx < CHP * HID; idx += 256) {
    int c = idx >> 10;          // 0..15
    int k = idx & (HID - 1);    // 0..1023
    ldsW[k * CHP + c] = (c < CH) ? W[c * HID + k] : 0.0f;
  }
  __syncthreads();

  const int wave = threadIdx.x >> 5;
  const int lane = threadIdx.x & 31;
  const int hi   = lane >> 4;     // 0: lanes 0-15, 1: lanes 16-31
  const int lo   = lane & 15;
  const int kh   = 2 * hi;        // K sub-offset within a K=4 step

  const int tile = blockIdx.x * 8 + wave;   // 2048 tiles total
  const int tok0 = tile * 16;

  // A-fragment source: row of this lane's token, starting at column kh.
  const float* aRow = x + (size_t)(tok0 + lo) * HID + kh;
  // B-fragment source in LDS: W[n = lo][k] stored at ldsW[k*16 + lo].
  const float* bBase = &ldsW[kh * CHP + lo];

  v8f acc = {};
#pragma unroll 8
  for (int k0 = 0; k0 < HID; k0 += 4) {
    if ((k0 & 63) == 0) {
      // keep the x stream ahead in cache (global_prefetch_b8)
      __builtin_prefetch(aRow + k0 + 256, 0, 0);
    }
    v2f a = *(const v2f*)(aRow + k0);
    float b0 = bBase[(k0 + 0) * CHP];
    float b1 = bBase[(k0 + 1) * CHP];
    v2f b = {b0, b1};
    acc = __builtin_amdgcn_wmma_f32_16x16x4_f32(
        /*neg_a=*/false, a, /*neg_b=*/false, b,
        /*c_mod=*/(short)0, acc, /*reuse_a=*/false, /*reuse_b=*/false);
  }

  // Batch index is uniform across the 16-token tile (16 | SEQ).
  const int bidx = tok0 >> 11;            // tok / SEQ
  const int* widTile = wid + tok0;

  // One count per token (lanes 0-15 each own one token of the tile).
  if (hi == 0) {
    int seg = bidx * NW + widTile[lo];
    atomicAdd(&counts[seg], 1.0f);
  }

  // Scatter D tile: lane holds channel n = lo for tokens M = r + 8*hi.
  const int n = lo;
#pragma unroll
  for (int r = 0; r < 8; ++r) {
    int m = r + 8 * hi;                   // token within tile
    int seg = bidx * NW + widTile[m];
    if (n < CH) {
      atomicAdd(&sums[seg * CHP + n], acc[r]);
    }
  }
}

// ---------------------------------------------------------------------------
// Kernel 2: per-token gather of segment mean + bias -> logits [B,S,15].
// ---------------------------------------------------------------------------
__global__ __launch_bounds__(256) void finalize_kernel(
    const int*   __restrict__ wid,
    const float* __restrict__ sums,
    const float* __restrict__ counts,
    const float* __restrict__ bias,
    float* __restrict__ out) {
  int t = blockIdx.x * 256 + threadIdx.x;
  if (t >= NTOK) return;
  int b   = t >> 11;                       // t / SEQ
  int seg = b * NW + wid[t];
  float inv = 1.0f / fmaxf(counts[seg], 1.0f);
  const float* sp = &sums[seg * CHP];
  float* op = out + (size_t)t * CH;
#pragma unroll
  for (int c = 0; c < CH; ++c) {
    op[c] = sp[c] * inv + bias[c];
  }
}

// ---------------------------------------------------------------------------
extern "C" void kernel_launch(void* const* d_in, const int* in_sizes, int n_in,
                              void* d_out, int out_size, void* d_ws, size_t ws_size,
                              hipStream_t stream) {
  const float* x    = (const float*)d_in[0];   // [16,2048,1024] fp32
  const int*   wid  = (const int*)d_in[1];     // [16,2048] int32
  const float* W    = (const float*)d_in[2];   // [15,1024] fp32
  const float* bias = (const float*)d_in[3];   // [15] fp32
  float* out = (float*)d_out;                  // [16,2048,15] fp32

  float* sums   = (float*)d_ws;                // NSEG*CHP floats
  float* counts = sums + (size_t)NSEG * CHP;   // NSEG floats

  zero_ws_kernel<<<(NSEG * CHP + 255) / 256, 256, 0, stream>>>(sums, counts);

  // 2048 tiles / 8 waves per block = 256 blocks
  gemm_scatter_kernel<<<(NTOK / 16) / 8, 256, 0, stream>>>(x, wid, W, sums, counts);

  finalize_kernel<<<(NTOK + 255) / 256, 256, 0, stream>>>(wid, sums, counts, bias, out);
}